// FLIMAdaptiveDecoderLayer_55396488184486
// MI455X (gfx1250) — compile-verified
//
#include <hip/hip_runtime.h>
#include <cstdint>

#define HWPX 65536
#define NC 64
#define NIMG 8
#define NBIN 256

typedef float v2f __attribute__((ext_vector_type(2)));
typedef float v8f __attribute__((ext_vector_type(8)));

// ws layout (float indices):
// scale[512]@0  bias[512]@512  meanf[512]@1024  fmn[512]@1536  fmx[512]@2048
// low[8]@2560  high[8]@2568  otsu[512]@2576  pfg[512]@3088  wgt[512]@3600
// outMin(u32)[64]@4112  outMax(u32)[64]@4176      (total 4240 floats ~17KB)

__global__ void k1_band_stats(const float* __restrict__ x, float* __restrict__ ws) {
    int band = blockIdx.x;                 // 0..511  (img*64 + c)
    int tid  = threadIdx.x;                // 256 threads
    const float* p = x + (size_t)band * HWPX;
    float mn = __builtin_inff(), mx = -__builtin_inff(), sm = 0.f;
    for (int i = tid; i < HWPX; i += 256) {
        float v = p[i];
        mn = fminf(mn, v); mx = fmaxf(mx, v); sm += v;
    }
    __shared__ float smn[256], smx[256], ssm[256];
    smn[tid] = mn; smx[tid] = mx; ssm[tid] = sm;
    __syncthreads();
    for (int s = 128; s > 0; s >>= 1) {
        if (tid < s) {
            smn[tid] = fminf(smn[tid], smn[tid + s]);
            smx[tid] = fmaxf(smx[tid], smx[tid + s]);
            ssm[tid] += ssm[tid + s];
        }
        __syncthreads();
    }
    if (tid == 0) {
        float bmn = smn[0], bmx = smx[0];
        float mean = ssm[0] * (1.f / HWPX);
        float rng = bmx - bmn;
        float sc, bi;
        if (rng != 0.f)      { sc = 1.f / rng; bi = bmn; }   // (x-mn)/rng
        else if (bmx != 0.f) { sc = 1.f / bmx; bi = bmn; }   // (x-mn)/mx
        else                 { sc = 1.f;       bi = 0.f; }   // x
        float f1 = (bmn - bi) * sc, f2 = (bmx - bi) * sc;
        ws[band]          = sc;
        ws[512 + band]    = bi;
        ws[1024 + band]   = (mean - bi) * sc;     // mean of normalized band
        ws[1536 + band]   = fminf(f1, f2);        // normalized min
        ws[2048 + band]   = fmaxf(f1, f2);        // normalized max
    }
}

// Otsu over the 64 per-band means of one image + variance -> low/high ends.
__global__ void k2_thresholds(float* __restrict__ ws) {
    if (threadIdx.x != 0) return;
    int b = blockIdx.x;                    // 0..7
    const float* m = ws + 1024 + b * NC;
    __shared__ float hist[NBIN], w2s[NBIN], s2s[NBIN];
    float mn = __builtin_inff(), mx = -__builtin_inff(), sm = 0.f;
    for (int c = 0; c < NC; c++) { float v = m[c]; mn = fminf(mn, v); mx = fmaxf(mx, v); sm += v; }
    float mean = sm / (float)NC;
    float var = 0.f;
    for (int c = 0; c < NC; c++) { float d = m[c] - mean; var += d * d; }
    var /= (float)NC;                      // ddof=0 (reference's "stdev" is variance)
    float width = (mx - mn) / (float)NBIN;
    float sw = (width > 0.f) ? width : 1.f;
    for (int k = 0; k < NBIN; k++) hist[k] = 0.f;
    for (int c = 0; c < NC; c++) {
        int idx = (int)floorf((m[c] - mn) / sw);
        idx = idx < 0 ? 0 : (idx > NBIN - 1 ? NBIN - 1 : idx);
        hist[idx] += 1.f;
    }
    float w2 = 0.f, s2 = 0.f;
    for (int k = NBIN - 1; k >= 0; k--) {
        float bc = mn + ((float)k + 0.5f) * width;
        w2 += hist[k]; s2 += hist[k] * bc;
        w2s[k] = w2; s2s[k] = s2;
    }
    float w1 = 0.f, s1 = 0.f, best = -1.f; int bidx = 0;
    for (int i = 0; i < NBIN - 1; i++) {
        float bc = mn + ((float)i + 0.5f) * width;
        w1 += hist[i]; s1 += hist[i] * bc;
        float m1 = s1 / fmaxf(w1, 1e-12f);
        float m2 = s2s[i + 1] / fmaxf(w2s[i + 1], 1e-12f);
        float d = m1 - m2;
        float v = w1 * w2s[i + 1] * d * d;
        if (v > best) { best = v; bidx = i; }
    }
    float thr = mn + ((float)bidx + 0.5f) * width;
    ws[2560 + b] = (thr > 0.1f) ? (thr - var) : thr;   // low_end
    ws[2568 + b] = thr + var;                          // high_end
}

// Per-band Otsu (256-bin) + foreground proportion.
__global__ void k3_band_otsu(const float* __restrict__ x, float* __restrict__ ws) {
    int band = blockIdx.x, tid = threadIdx.x;
    const float* p = x + (size_t)band * HWPX;
    float sc  = ws[band],        bi  = ws[512 + band];
    float fmn = ws[1536 + band], fmx = ws[2048 + band];
    float width = (fmx - fmn) * (1.f / (float)NBIN);
    float sw = (width > 0.f) ? width : 1.f;
    float inv_sw = 1.f / sw;
    __shared__ unsigned hist[NBIN];
    __shared__ float w2s[NBIN], s2s[NBIN];
    __shared__ float thr_sh;
    __shared__ unsigned ctot;
    hist[tid] = 0u;
    if (tid == 0) ctot = 0u;
    __syncthreads();
    for (int i = tid; i < HWPX; i += 256) {
        float f = (p[i] - bi) * sc;
        int idx = (int)floorf((f - fmn) * inv_sw);
        idx = idx < 0 ? 0 : (idx > NBIN - 1 ? NBIN - 1 : idx);
        atomicAdd(&hist[idx], 1u);
    }
    __syncthreads();
    if (tid == 0) {
        float w2 = 0.f, s2 = 0.f;
        for (int k = NBIN - 1; k >= 0; k--) {
            float h = (float)hist[k];
            float bc = fmn + ((float)k + 0.5f) * width;
            w2 += h; s2 += h * bc;
            w2s[k] = w2; s2s[k] = s2;
        }
        float w1 = 0.f, s1 = 0.f, best = -1.f; int bidx = 0;
        for (int i = 0; i < NBIN - 1; i++) {
            float h = (float)hist[i];
            float bc = fmn + ((float)i + 0.5f) * width;
            w1 += h; s1 += h * bc;
            float m1 = s1 / fmaxf(w1, 1e-12f);
            float m2 = s2s[i + 1] / fmaxf(w2s[i + 1], 1e-12f);
            float d = m1 - m2;
            float v = w1 * w2s[i + 1] * d * d;
            if (v > best) { best = v; bidx = i; }
        }
        thr_sh = fmn + ((float)bidx + 0.5f) * width;
    }
    __syncthreads();
    float thr = thr_sh;
    unsigned cnt = 0u;
    for (int i = tid; i < HWPX; i += 256) {
        float f = (p[i] - bi) * sc;
        cnt += (f > thr) ? 1u : 0u;
    }
    atomicAdd(&ctot, cnt);
    __syncthreads();
    if (tid == 0) {
        ws[2576 + band] = thr;
        ws[3088 + band] = (float)ctot * (1.f / (float)HWPX);
    }
}

// Weight matrix + init output plane min/max accumulators.
__global__ void k5_weights(float* __restrict__ ws) {
    int t = blockIdx.x * 256 + threadIdx.x;     // 0..511
    if (t < 512) {
        int b = t >> 6;
        float m = ws[1024 + t], p = ws[3088 + t];
        float low = ws[2560 + b], high = ws[2568 + b];
        float wv = (m >= high && p > 0.2f) ? -1.f
                 : ((m <= low && p < 0.1f) ? 1.f : 0.f);
        ws[3600 + t] = wv;
    }
    if (t < 64) {
        ((unsigned*)ws)[4112 + t] = 0x7f800000u;  // +inf
        ((unsigned*)ws)[4176 + t] = 0u;           // 0.0f (ReLU'd output >= 0)
    }
}

// WMMA einsum: y[i,n,hw] = relu(255 * sum_c feat[i,c,hw]*w[n,c]); track plane min/max.
// A (16x4) is w[n,c] physically zero-padded to 16 rows in LDS -> unconditional
// ds_load_b64 fragments, no EXEC-predicated loads around the WMMAs.
__global__ void k6_einsum_wmma(const float* __restrict__ x, float* __restrict__ out,
                               float* __restrict__ ws) {
    __shared__ float wl[1024];                // 16 rows x 64; rows 8..15 = 0 padding
    __shared__ float scl[NC], bil[NC];
    __shared__ unsigned lmin[NIMG], lmax[NIMG];
    int tid  = threadIdx.x;
    int img  = blockIdx.x >> 9;               // 8 images
    int tile = blockIdx.x & 511;              // 512 tiles of 128 px
    wl[tid]       = ws[3600 + tid];
    wl[tid + 256] = ws[3600 + tid + 256];
    wl[tid + 512] = 0.f;                      // zero-pad rows 8..15
    wl[tid + 768] = 0.f;
    if (tid < NC) { scl[tid] = ws[img * NC + tid]; bil[tid] = ws[512 + img * NC + tid]; }
    if (tid < NIMG) { lmin[tid] = 0x7f800000u; lmax[tid] = 0u; }
    __syncthreads();

    int lane = tid & 31, wave = tid >> 5;
    int row  = lane & 15, half = lane >> 4;   // A/B K-split across lane halves
    int hw   = tile * 128 + wave * 16 + row;
    const float* base = x + (size_t)img * NC * HWPX;

    v8f acc = {0.f, 0.f, 0.f, 0.f, 0.f, 0.f, 0.f, 0.f};
    for (int k4 = 0; k4 < NC; k4 += 4) {
        int ka = k4 + half * 2;
        v2f a, bm;
        // A (16x4 f32): lanes 0-15 hold K={k4,k4+1}, lanes 16-31 K={k4+2,k4+3}
        a.x = wl[row * NC + ka];
        a.y = wl[row * NC + ka + 1];
        // B (4x16 f32): same K split; N = hw column; normalize on the fly
        bm.x = (base[(size_t)ka * HWPX + hw]       - bil[ka])     * scl[ka];
        bm.y = (base[(size_t)(ka + 1) * HWPX + hw] - bil[ka + 1]) * scl[ka + 1];
        acc = __builtin_amdgcn_wmma_f32_16x16x4_f32(
            /*neg_a=*/false, a, /*neg_b=*/false, bm,
            /*c_mod=*/(short)0, acc, /*reuse_a=*/false, /*reuse_b=*/false);
    }
    // D rows 0..7 (the valid n's) live in VGPRs 0..7 of lanes 0..15
    if (half == 0) {
        for (int r = 0; r < NIMG; r++) {
            float v = fmaxf(acc[r] * 255.f, 0.f);
            out[((size_t)(img * NIMG + r)) * HWPX + hw] = v;
            unsigned u = __float_as_uint(v);
            atomicMin(&lmin[r], u);
            atomicMax(&lmax[r], u);
        }
    }
    __syncthreads();
    if (tid < NIMG) {
        atomicMin(&((unsigned*)ws)[4112 + img * NIMG + tid], lmin[tid]);
        atomicMax(&((unsigned*)ws)[4176 + img * NIMG + tid], lmax[tid]);
    }
}

// In-place per-plane min-max normalize * 255.
__global__ void k7_norm(float* __restrict__ out, const float* __restrict__ ws) {
    size_t idx = ((size_t)blockIdx.x * 256 + threadIdx.x) * 4;
    int plane = (int)(idx >> 16);
    float mn = __uint_as_float(((const unsigned*)ws)[4112 + plane]);
    float mx = __uint_as_float(((const unsigned*)ws)[4176 + plane]);
    float rng = mx - mn;
    float4 v = *(float4*)(out + idx);
    float* c = &v.x;
    #pragma unroll
    for (int j = 0; j < 4; j++) {
        float y = c[j];
        if (rng != 0.f)      y = (y - mn) / rng;
        else if (mx != 0.f)  y = (y - mn) / mx;
        c[j] = y * 255.f;
    }
    *(float4*)(out + idx) = v;
}

extern "C" void kernel_launch(void* const* d_in, const int* in_sizes, int n_in,
                              void* d_out, int out_size, void* d_ws, size_t ws_size,
                              hipStream_t stream) {
    const float* x = (const float*)d_in[0];   // [8,64,256,256] f32
    float* out = (float*)d_out;               // [8,8,256,256]  f32
    float* ws  = (float*)d_ws;

    k1_band_stats<<<512, 256, 0, stream>>>(x, ws);
    k2_thresholds<<<NIMG, 32, 0, stream>>>(ws);
    k3_band_otsu<<<512, 256, 0, stream>>>(x, ws);
    k5_weights<<<2, 256, 0, stream>>>(ws);
    k6_einsum_wmma<<<NIMG * 512, 256, 0, stream>>>(x, out, ws);
    k7_norm<<<4096, 256, 0, stream>>>(out, ws);
}